// VGAE_73392401154214
// MI455X (gfx1250) — compile-verified
//
#include <hip/hip_runtime.h>
#include <math.h>

#define N_NODES 100000
#define N_EDGES 3200000
#define F_IN    512
#define HID     32
#define F_OUT   16

typedef __attribute__((ext_vector_type(2))) float v2f;
typedef __attribute__((ext_vector_type(8))) float v8f;

__device__ __forceinline__ void atom_add_f32(float* p, float v) {
    // lowers to global_atomic_add_f32 (hardware L2 fadd) on gfx1250
    unsafeAtomicAdd(p, v);
}

// ---------------------------------------------------------------------------
// Zero-fill (grid-stride)
// ---------------------------------------------------------------------------
__global__ void vgae_zero_kernel(float* __restrict__ p, long long n) {
    long long i = (long long)blockIdx.x * blockDim.x + threadIdx.x;
    long long stride = (long long)gridDim.x * blockDim.x;
    for (; i < n; i += stride) p[i] = 0.0f;
}

// ---------------------------------------------------------------------------
// GEMM1: hpre[N,32] = x[N,512] @ W1[512,32]   (fp32 WMMA 16x16x4)
// W1 staged in LDS (64KB). 256 threads = 8 waves; wave w owns rows
// blockIdx.x*128 + w*16, producing a 16x32 tile with two accumulators.
// A layout (ISA 7.12.2, 32-bit A 16x4): lane m=l&15 is row M, lane bit4
// selects K pair {0,1} vs {2,3}; VGPR0=K, VGPR1=K+1.  B (4x16) mirrored:
// lanes 0-15 hold K,K+1 for N=lane; lanes 16-31 hold K+2,K+3.
// ---------------------------------------------------------------------------
__global__ __launch_bounds__(256) void vgae_gemm1_kernel(
        const float* __restrict__ x, const float* __restrict__ W1,
        float* __restrict__ hpre) {
    __shared__ float lW[F_IN * HID];              // 64 KB
    {
        const float4* Wv = (const float4*)W1;
        float4* lWv = (float4*)lW;
        for (int i = threadIdx.x; i < F_IN * HID / 4; i += 256) lWv[i] = Wv[i];
    }
    __syncthreads();

    const int wave = threadIdx.x >> 5;
    const int lane = threadIdx.x & 31;
    const int m    = lane & 15;
    const int kh   = lane >> 4;                    // 0 or 1
    const int row0 = blockIdx.x * 128 + wave * 16;

    int arow = row0 + m;
    if (arow >= N_NODES) arow = N_NODES - 1;       // clamp (stores are guarded)
    const float* xrow = x + (size_t)arow * F_IN;

    v8f acc0 = {};
    v8f acc1 = {};
    #pragma unroll 4
    for (int kb = 0; kb < F_IN; kb += 4) {
        const int k = kb + kh * 2;
        v2f a;  a.x  = xrow[k];               a.y  = xrow[k + 1];
        v2f b0; b0.x = lW[k * HID + m];       b0.y = lW[(k + 1) * HID + m];
        v2f b1; b1.x = lW[k * HID + 16 + m];  b1.y = lW[(k + 1) * HID + 16 + m];
        acc0 = __builtin_amdgcn_wmma_f32_16x16x4_f32(false, a, false, b0,
                                                     (short)0, acc0, false, false);
        acc1 = __builtin_amdgcn_wmma_f32_16x16x4_f32(false, a, false, b1,
                                                     (short)0, acc1, false, false);
    }

    // D layout: VGPR r, lanes 0-15 -> M=r, lanes 16-31 -> M=r+8; N = m
    #pragma unroll
    for (int r = 0; r < 8; ++r) {
        const int orow = row0 + r + kh * 8;
        if (orow < N_NODES) {
            hpre[(size_t)orow * HID + m]      = acc0[r];
            hpre[(size_t)orow * HID + 16 + m] = acc1[r];
        }
    }
}

// ---------------------------------------------------------------------------
// SPMM (d=32): agg[dst] += w * feat[src].  8 lanes per edge, float4 per lane.
// ---------------------------------------------------------------------------
__global__ __launch_bounds__(256) void vgae_spmm32_kernel(
        const int* __restrict__ esrc, const int* __restrict__ edst,
        const float* __restrict__ ew, const float* __restrict__ feat,
        float* __restrict__ out) {
    const long long t = (long long)blockIdx.x * blockDim.x + threadIdx.x;
    const long long e = t >> 3;
    if (e >= N_EDGES) return;
    const int c = ((int)t & 7) * 4;
    const int s = esrc[e];
    const int d = edst[e];
    const float w = ew[e];
    const float4 v = *(const float4*)(feat + (size_t)s * 32 + c);
    float* op = out + (size_t)d * 32 + c;
    atom_add_f32(op + 0, w * v.x);
    atom_add_f32(op + 1, w * v.y);
    atom_add_f32(op + 2, w * v.z);
    atom_add_f32(op + 3, w * v.w);
}

// ---------------------------------------------------------------------------
// GEMM2: t[N,32] = relu(agg[N,32]) @ [Wmu | Wls] (32x32), fp32 WMMA
// ---------------------------------------------------------------------------
__global__ __launch_bounds__(256) void vgae_gemm2_kernel(
        const float* __restrict__ agg, const float* __restrict__ Wmu,
        const float* __restrict__ Wls, float* __restrict__ tbuf) {
    __shared__ float lW[HID * 32];                 // 4 KB, [k*32 + c]
    for (int i = threadIdx.x; i < HID * 32; i += 256) {
        const int k = i >> 5, c = i & 31;
        lW[i] = (c < 16) ? Wmu[k * 16 + c] : Wls[k * 16 + (c - 16)];
    }
    __syncthreads();

    const int wave = threadIdx.x >> 5;
    const int lane = threadIdx.x & 31;
    const int m    = lane & 15;
    const int kh   = lane >> 4;
    const int row0 = blockIdx.x * 128 + wave * 16;

    int arow = row0 + m;
    if (arow >= N_NODES) arow = N_NODES - 1;
    const float* hrow = agg + (size_t)arow * HID;

    v8f acc0 = {};
    v8f acc1 = {};
    #pragma unroll
    for (int kb = 0; kb < HID; kb += 4) {
        const int k = kb + kh * 2;
        v2f a;  a.x  = fmaxf(hrow[k], 0.0f);  a.y  = fmaxf(hrow[k + 1], 0.0f);
        v2f b0; b0.x = lW[k * 32 + m];        b0.y = lW[(k + 1) * 32 + m];
        v2f b1; b1.x = lW[k * 32 + 16 + m];   b1.y = lW[(k + 1) * 32 + 16 + m];
        acc0 = __builtin_amdgcn_wmma_f32_16x16x4_f32(false, a, false, b0,
                                                     (short)0, acc0, false, false);
        acc1 = __builtin_amdgcn_wmma_f32_16x16x4_f32(false, a, false, b1,
                                                     (short)0, acc1, false, false);
    }

    #pragma unroll
    for (int r = 0; r < 8; ++r) {
        const int orow = row0 + r + kh * 8;
        if (orow < N_NODES) {
            tbuf[(size_t)orow * 32 + m]      = acc0[r];
            tbuf[(size_t)orow * 32 + 16 + m] = acc1[r];
        }
    }
}

// ---------------------------------------------------------------------------
// SPMM (split): columns 0-15 of t -> mu, 16-31 -> logstd.  One edge sweep.
// ---------------------------------------------------------------------------
__global__ __launch_bounds__(256) void vgae_spmm_split_kernel(
        const int* __restrict__ esrc, const int* __restrict__ edst,
        const float* __restrict__ ew, const float* __restrict__ tbuf,
        float* __restrict__ mu, float* __restrict__ ls) {
    const long long t = (long long)blockIdx.x * blockDim.x + threadIdx.x;
    const long long e = t >> 3;
    if (e >= N_EDGES) return;
    const int c = ((int)t & 7) * 4;
    const int s = esrc[e];
    const int d = edst[e];
    const float w = ew[e];
    const float4 v = *(const float4*)(tbuf + (size_t)s * 32 + c);
    float* op = (c < 16) ? (mu + (size_t)d * F_OUT + c)
                         : (ls + (size_t)d * F_OUT + (c - 16));
    atom_add_f32(op + 0, w * v.x);
    atom_add_f32(op + 1, w * v.y);
    atom_add_f32(op + 2, w * v.z);
    atom_add_f32(op + 3, w * v.w);
}

// ---------------------------------------------------------------------------
// z = eps * exp(logstd) + mu      (mu/logstd already live in d_out)
// ---------------------------------------------------------------------------
__global__ __launch_bounds__(256) void vgae_repar_kernel(
        const float* __restrict__ eps, float* __restrict__ out) {
    const long long n = (long long)N_NODES * F_OUT;
    const long long i = (long long)blockIdx.x * blockDim.x + threadIdx.x;
    if (i >= n) return;
    const float mu = out[n + i];
    const float lg = out[2 * n + i];
    out[i] = eps[i] * expf(lg) + mu;
}

// ---------------------------------------------------------------------------
extern "C" void kernel_launch(void* const* d_in, const int* in_sizes, int n_in,
                              void* d_out, int out_size, void* d_ws, size_t ws_size,
                              hipStream_t stream) {
    const float* x    = (const float*)d_in[0];
    const int*   esrc = (const int*)  d_in[1];
    const int*   edst = (const int*)  d_in[2];
    const float* ew   = (const float*)d_in[3];
    const float* eps  = (const float*)d_in[4];
    const float* W1   = (const float*)d_in[5];
    const float* Wmu  = (const float*)d_in[6];
    const float* Wls  = (const float*)d_in[7];
    float* out = (float*)d_out;

    float* hpre = (float*)d_ws;                         // N*32 floats
    float* agg  = hpre + (size_t)N_NODES * HID;         // N*32 floats
    float* tbuf = hpre;                                 // reuse after spmm1
    float* mu   = out + (size_t)N_NODES * F_OUT;
    float* ls   = out + 2 * (size_t)N_NODES * F_OUT;

    const int gemm_blocks = (N_NODES + 127) / 128;              // 782
    const int edge_blocks = (int)(((long long)N_EDGES * 8) / 256); // 100000

    // 1) zero accumulation targets
    vgae_zero_kernel<<<2048, 256, 0, stream>>>(agg, (long long)N_NODES * HID);
    vgae_zero_kernel<<<2048, 256, 0, stream>>>(mu,  (long long)N_NODES * F_OUT * 2);

    // 2) hpre = x @ W1
    vgae_gemm1_kernel<<<gemm_blocks, 256, 0, stream>>>(x, W1, hpre);

    // 3) agg = A @ hpre
    vgae_spmm32_kernel<<<edge_blocks, 256, 0, stream>>>(esrc, edst, ew, hpre, agg);

    // 4) tbuf = relu(agg) @ [Wmu|Wls]
    vgae_gemm2_kernel<<<gemm_blocks, 256, 0, stream>>>(agg, Wmu, Wls, tbuf);

    // 5) mu = A @ tbuf[:, :16], logstd = A @ tbuf[:, 16:]
    vgae_spmm_split_kernel<<<edge_blocks, 256, 0, stream>>>(esrc, edst, ew, tbuf, mu, ls);

    // 6) z = eps * exp(logstd) + mu
    const int n_out = N_NODES * F_OUT;
    vgae_repar_kernel<<<(n_out + 255) / 256, 256, 0, stream>>>(eps, out);
}